// HMM_54443005444853
// MI455X (gfx1250) — compile-verified
//
#include <hip/hip_runtime.h>

// HMM posterior smoothing: B=32, T=2048, K=512, f32.
// Pass 1: per-chunk emission sums.  Pass 2 (fused): chunk prefix -> carry/total.
// Main: WMMA-based intra-tile cumsum (cs = L x E + carry, fp32 16x16x4 WMMA),
//       DPP ROW_XMASK butterfly LSE over K, streamed 134 MB store
//       (~5.8 us floor @ 23.3 TB/s).

#define Bn 32
#define Tn 2048
#define Kn 512
#define NCHUNK 16
#define TCHUNK 128            // Tn / NCHUNK
#define HALF_LOG_2PI 0.9189385332046727f

typedef float v2f __attribute__((ext_vector_type(2)));
typedef float v8f __attribute__((ext_vector_type(8)));

// Lane-XOR shuffle via DPP16 ROW_XMASK (0x160 | mask): pure VALU, no LDS wait.
// Masks <= 8 stay inside each 16-lane row, so the two half-wave rows (t = r and
// t = r+8) reduce independently.
template <int MASK>
__device__ __forceinline__ float dpp_xor(float v) {
  return __int_as_float(__builtin_amdgcn_update_dpp(
      0, __float_as_int(v), 0x160 | MASK, 0xf, 0xf, true));
}

__device__ __forceinline__ float red_max16(float v) {
  v = fmaxf(v, dpp_xor<1>(v));
  v = fmaxf(v, dpp_xor<2>(v));
  v = fmaxf(v, dpp_xor<4>(v));
  v = fmaxf(v, dpp_xor<8>(v));
  return v;
}

__device__ __forceinline__ float red_sum16(float v) {
  v += dpp_xor<1>(v);
  v += dpp_xor<2>(v);
  v += dpp_xor<4>(v);
  v += dpp_xor<8>(v);
  return v;
}

// ---------------- Pass 1: chunk partial sums of ln_emis over t ----------------
__global__ void __launch_bounds__(Kn) hmm_chunksum(
    const float* __restrict__ obvs,
    const float* __restrict__ mu,
    const float* __restrict__ log_sigma,
    float* __restrict__ chunk_sum)   // [Bn][NCHUNK][Kn]
{
  __shared__ float xs[TCHUNK];
  const int blk = blockIdx.x;
  const int b   = blk / NCHUNK;
  const int c   = blk % NCHUNK;
  const int tid = threadIdx.x;
  if (tid < TCHUNK) xs[tid] = obvs[b * Tn + c * TCHUNK + tid];
  __syncthreads();

  const int   k   = tid;
  const float ls  = log_sigma[k];
  const float is  = expf(-ls);
  const float mis = mu[k] * is;
  const float c2  = -ls - HALF_LOG_2PI;

  float acc = 0.0f;
#pragma unroll 8
  for (int i = 0; i < TCHUNK; ++i) {
    float z = fmaf(xs[i], is, -mis);
    acc += fmaf(z, -0.5f * z, c2);
  }
  chunk_sum[blk * Kn + k] = acc;
}

// ---------------- Main: WMMA cumsum + log_gamma + logsumexp + store ----------
__global__ void __launch_bounds__(Kn) hmm_main(
    const float* __restrict__ obvs,
    const float* __restrict__ mu,
    const float* __restrict__ log_sigma,
    const float* __restrict__ ln_pi,
    const float* __restrict__ ln_diag,
    const float* __restrict__ chunk_sum,
    float* __restrict__ out)
{
  __shared__ float xs[TCHUNK];
  __shared__ float pmax[16 * 16];   // [wave][row]
  __shared__ float psum[16 * 16];
  __shared__ float lse_s[16];

  const int  blk  = blockIdx.x;
  const int  b    = blk / NCHUNK;
  const int  c    = blk % NCHUNK;
  const int  tid  = threadIdx.x;
  const int  wave = tid >> 5;
  const int  lane = tid & 31;
  const int  ln   = lane & 15;
  const bool hi   = lane >= 16;

  if (tid < TCHUNK) xs[tid] = obvs[b * Tn + c * TCHUNK + tid];

  // Each wave owns states [wave*32, wave*32+32): two 16-wide k-tiles.
  const int kA = wave * 32 + ln;
  const int kB = kA + 16;

  const float lsA  = log_sigma[kA], lsB  = log_sigma[kB];
  const float isA  = expf(-lsA),    isB  = expf(-lsB);
  const float misA = mu[kA] * isA,  misB = mu[kB] * isB;
  const float c2A  = -lsA - HALF_LOG_2PI, c2B = -lsB - HALF_LOG_2PI;
  const float piA  = ln_pi[kA],   piB = ln_pi[kB];
  const float ldA  = ln_diag[kA], ldB = ln_diag[kB];

  // Sequential prefix over the 16 chunk sums -> carry into this chunk + total.
  float carryA = 0.f, carryB = 0.f, totA = 0.f, totB = 0.f;
  const float* csp = chunk_sum + b * NCHUNK * Kn;
  for (int cc = 0; cc < NCHUNK; ++cc) {
    float sA = csp[cc * Kn + kA];
    float sB = csp[cc * Kn + kB];
    if (cc < c) { carryA += sA; carryB += sB; }
    totA += sA; totB += sB;
  }

  // ln_emis at t = T-1 (for ln_bs) and P2 = ln_pi + ln_emis_last.
  const float xl  = obvs[b * Tn + (Tn - 1)];
  const float zlA = fmaf(xl, isA, -misA);
  const float zlB = fmaf(xl, isB, -misB);
  const float P2A = piA + fmaf(zlA, -0.5f * zlA, c2A);
  const float P2B = piB + fmaf(zlB, -0.5f * zlB, c2B);

  // A operand = lower-triangular ones L[:,4j:4j+4].  16x4 f32 A layout:
  // lanes 0-15 and 16-31 both hold M = lane&15; VGPR0/VGPR1 hold cols {0,2}/{1,3}.
  const int M    = ln;
  const int koff = hi ? 2 : 0;
  v2f aL[4];
#pragma unroll
  for (int j = 0; j < 4; ++j) {
    aL[j].x = (4 * j + koff     <= M) ? 1.0f : 0.0f;
    aL[j].y = (4 * j + koff + 1 <= M) ? 1.0f : 0.0f;
  }

  const int   toff = hi ? 2 : 0;   // B operand rows (t offsets) this half supplies
  const int   mrow = hi ? 8 : 0;   // C/D row offset for this half
  const float Tm1  = (float)(Tn - 1);
  float* outp = out + (size_t)b * Tn * Kn;

  __syncthreads();

  for (int tt = 0; tt < TCHUNK / 16; ++tt) {
    const int t0 = c * TCHUNK + tt * 16;

    // C = running carry broadcast to every row.
    v8f accA, accB;
#pragma unroll
    for (int r = 0; r < 8; ++r) { accA[r] = carryA; accB[r] = carryB; }

    // cs(16t x 16k) = sum_j L[:,4j:4j+4] x E[4j:4j+4,:] + carry   (4 chained WMMAs)
#pragma unroll
    for (int j = 0; j < 4; ++j) {
      const int   xi = tt * 16 + 4 * j + toff;
      const float x0 = xs[xi];
      const float x1 = xs[xi + 1];
      float zA0 = fmaf(x0, isA, -misA), zA1 = fmaf(x1, isA, -misA);
      float zB0 = fmaf(x0, isB, -misB), zB1 = fmaf(x1, isB, -misB);
      v2f bA, bB;
      bA.x = fmaf(zA0, -0.5f * zA0, c2A);
      bA.y = fmaf(zA1, -0.5f * zA1, c2A);
      bB.x = fmaf(zB0, -0.5f * zB0, c2B);
      bB.y = fmaf(zB1, -0.5f * zB1, c2B);
      accA = __builtin_amdgcn_wmma_f32_16x16x4_f32(false, aL[j], false, bA,
                                                   (short)0, accA, false, false);
      accB = __builtin_amdgcn_wmma_f32_16x16x4_f32(false, aL[j], false, bB,
                                                   (short)0, accB, false, false);
    }

    // New carry = cs row M=15 (VGPR 7, lanes 16-31); broadcast to both halves.
    const float cnA = __shfl(accA[7], ln + 16, 32);
    const float cnB = __shfl(accB[7], ln + 16, 32);

    float lgA[8], lgB[8];
#pragma unroll
    for (int r = 0; r < 8; ++r) {
      const float tf  = (float)(t0 + r + mrow);
      const float csA = accA[r];
      const float csB = accB[r];
      // ln_fs + ln_bs, mirroring the reference's add order.
      lgA[r] = ((csA + piA) + tf * ldA) + ((P2A + (totA - csA)) + (Tm1 - tf) * ldA);
      lgB[r] = ((csB + piB) + tf * ldB) + ((P2B + (totB - csB)) + (Tm1 - tf) * ldB);

      // Partial LSE over this wave's 32 states: DPP butterfly within each
      // 16-lane row (rows r and r+8 reduce independently).
      float m = red_max16(fmaxf(lgA[r], lgB[r]));
      float s = red_sum16(expf(lgA[r] - m) + expf(lgB[r] - m));
      if (ln == 0) {
        pmax[wave * 16 + r + mrow] = m;
        psum[wave * 16 + r + mrow] = s;
      }
    }
    __syncthreads();

    // Combine the 16 wave-partials per timestep row.
    if (tid < 16) {
      const int row = tid;
      float gm = pmax[row];
#pragma unroll
      for (int w = 1; w < 16; ++w) gm = fmaxf(gm, pmax[w * 16 + row]);
      float gs = 0.0f;
#pragma unroll
      for (int w = 0; w < 16; ++w) gs += psum[w * 16 + row] * expf(pmax[w * 16 + row] - gm);
      lse_s[row] = gm + logf(gs);
    }
    __syncthreads();

    // Normalize and stream out: per row, 16 consecutive floats per k-tile.
#pragma unroll
    for (int r = 0; r < 8; ++r) {
      const float  l = lse_s[r + mrow];
      const size_t o = (size_t)(t0 + r + mrow) * Kn;
      outp[o + kA] = lgA[r] - l;
      outp[o + kB] = lgB[r] - l;
    }
    carryA = cnA;
    carryB = cnB;
  }
}

extern "C" void kernel_launch(void* const* d_in, const int* in_sizes, int n_in,
                              void* d_out, int out_size, void* d_ws, size_t ws_size,
                              hipStream_t stream) {
  (void)in_sizes; (void)n_in; (void)out_size; (void)ws_size;
  const float* obvs      = (const float*)d_in[0];
  const float* mu        = (const float*)d_in[1];
  const float* log_sigma = (const float*)d_in[2];
  const float* ln_pi     = (const float*)d_in[3];
  const float* ln_diag   = (const float*)d_in[4];
  float* out       = (float*)d_out;
  float* chunk_sum = (float*)d_ws;   // Bn*NCHUNK*Kn floats = 1 MiB scratch

  dim3 grid(Bn * NCHUNK), block(Kn);
  hmm_chunksum<<<grid, block, 0, stream>>>(obvs, mu, log_sigma, chunk_sum);
  hmm_main<<<grid, block, 0, stream>>>(obvs, mu, log_sigma, ln_pi, ln_diag,
                                       chunk_sum, out);
}